// ConvLayersV2_45672682226530
// MI455X (gfx1250) — compile-verified
//
#include <hip/hip_runtime.h>

typedef float v2f __attribute__((ext_vector_type(2)));
typedef float v8f __attribute__((ext_vector_type(8)));

// ---------------------------------------------------------------------------
// Implicit-GEMM 3x3 VALID conv, stride 1, fp32, via V_WMMA_F32_16X16X4_F32.
// All tensor dimensions are template parameters so every in-loop address is
// either a load immediate or a literal pointer increment.
//
// Activations: channel-pair interleaved [N, C/2, H, W, 2] -> each lane's B
// operand (two consecutive K == two consecutive channels at one pixel) is one
// aligned global_load_b64 directly into the WMMA register pair.
// Weights: packed wp[o, tap, c] -> each lane's A operand is one b64 load at
// an immediate offset. Loop nest c-outer / tap-inner (all-immediate body).
//
// One wave: 16(oc) x 32(ow) = two 16x16 tiles sharing A (2 acc chains).
// A lane: row M = L%16 (oc), K pair (L/16)*2.   B lane: col N = L%16 (ow).
// D: VGPR r: lanes 0-15 -> M=r, lanes 16-31 -> M=r+8, N = L%16.
// ---------------------------------------------------------------------------
template <int CINP, int HIN, int WIN, int OCN, bool ILV_OUT>
__launch_bounds__(256)
__global__ void conv3x3_wmma_f32(const float* __restrict__ in,  // [N,CINP/2,HIN,WIN,2]
                                 const float* __restrict__ wp,  // [OCN,9,CINP]
                                 float* __restrict__ out,
                                 int totalWaves)
{
    constexpr int CP2 = CINP / 2;
    constexpr int HO = HIN - 2, WO = WIN - 2;
    constexpr int OWT = (WO + 31) / 32;          // pixel tiles per row (8: pow2)
    constexpr int OCT = OCN / 16;                // oc tiles (4 or 8: pow2)
    constexpr size_t CHAN2 = (size_t)HIN * WIN * 2;  // floats per channel-pair
    constexpr size_t HOWO  = (size_t)HO * WO;

    const int lane = threadIdx.x & 31;
    const int wave = (int)((blockIdx.x * blockDim.x + threadIdx.x) >> 5);
    if (wave >= totalWaves) return;              // whole-wave exit: EXEC full

    const int ocTile = wave % OCT;               // pow2 -> mask
    int pixTile      = wave / OCT;               // pow2 -> shift
    const int owT = pixTile % OWT;               // pow2 -> mask
    pixTile /= OWT;                              // pow2 -> shift
    const int oh = pixTile % HO;                 // one const-div per wave
    const int n  = pixTile / HO;

    const int oc0 = ocTile << 4;
    const int ow0 = owT << 5;

    const int half = lane >> 4;                  // K sub-pair selector
    const int l16  = lane & 15;

    // A: packed weight row; lane's pair starts at +half*2, contiguous in k.
    const float* __restrict__ aptr =
        wp + (size_t)(oc0 + l16) * (9 * CINP) + half * 2;

    // Pixel columns (clamped for ragged tail; garbage columns never stored).
    const int ow0l = ow0 + l16;
    const int ow1l = ow0 + 16 + l16;
    const int owc0 = (ow0l < WO) ? ow0l : (WO - 1);
    const int owc1 = (ow1l < WO) ? ow1l : (WO - 1);

    const float* __restrict__ b0 =
        in + (size_t)(n * CP2 + half) * CHAN2 + ((size_t)oh * WIN + owc0) * 2;
    const float* __restrict__ b1 =
        in + (size_t)(n * CP2 + half) * CHAN2 + ((size_t)oh * WIN + owc1) * 2;

    v8f acc0 = {0, 0, 0, 0, 0, 0, 0, 0};
    v8f acc1 = {0, 0, 0, 0, 0, 0, 0, 0};

    for (int c = 0; c < CINP; c += 4) {
#pragma unroll
        for (int tap = 0; tap < 9; ++tap) {
            constexpr int xx = 0;  (void)xx;
            const int kh = tap / 3, kw = tap % 3;          // constants
            const v2f av  = *(const v2f*)(aptr + tap * CINP);
            const v2f bv0 = *(const v2f*)(b0 + (kh * WIN + kw) * 2);
            const v2f bv1 = *(const v2f*)(b1 + (kh * WIN + kw) * 2);
            acc0 = __builtin_amdgcn_wmma_f32_16x16x4_f32(
                       false, av, false, bv0, (short)0, acc0, false, false);
            acc1 = __builtin_amdgcn_wmma_f32_16x16x4_f32(
                       false, av, false, bv1, (short)0, acc1, false, false);
        }
        aptr += 4;              // next 4-channel K block in packed weights
        b0 += 2 * CHAN2;        // advance two channel-pairs (literal stride)
        b1 += 2 * CHAN2;
    }

    if (ILV_OUT) {
        // Store interleaved [N, OCN/2, HO, WO, 2]: b64 per channel-pair.
        const int cp0 = (oc0 + half * 8) >> 1;
        float* __restrict__ ob =
            out + ((size_t)(n * (OCN / 2) + cp0) * HOWO + (size_t)oh * WO) * 2;
        if (ow0l < WO) {
#pragma unroll
            for (int r = 0; r < 4; ++r) {
                v2f v = {acc0[2 * r], acc0[2 * r + 1]};
                *(v2f*)(ob + (size_t)r * HOWO * 2 + (size_t)ow0l * 2) = v;
            }
        }
        if (ow1l < WO) {
#pragma unroll
            for (int r = 0; r < 4; ++r) {
                v2f v = {acc1[2 * r], acc1[2 * r + 1]};
                *(v2f*)(ob + (size_t)r * HOWO * 2 + (size_t)ow1l * 2) = v;
            }
        }
    } else {
        // Plain NCHW stores (final output, streamed once -> non-temporal).
        float* __restrict__ ob =
            out + ((size_t)n * OCN + (oc0 + half * 8)) * HOWO + (size_t)oh * WO;
        if (ow0l < WO) {
#pragma unroll
            for (int r = 0; r < 8; ++r)
                __builtin_nontemporal_store(acc0[r], ob + (size_t)r * HOWO + ow0l);
        }
        if (ow1l < WO) {
#pragma unroll
            for (int r = 0; r < 8; ++r)
                __builtin_nontemporal_store(acc1[r], ob + (size_t)r * HOWO + ow1l);
        }
    }
}

// wp[o, tap, c] = (c < C) ? w[o, c, kh, kw] : 0     (tap = kh*3+kw)
__global__ void pack_weights(const float* __restrict__ w, float* __restrict__ wp,
                             int O, int C, int CP)
{
    const int idx = blockIdx.x * blockDim.x + threadIdx.x;
    const int total = O * 9 * CP;
    if (idx >= total) return;
    const int c   = idx % CP;
    const int tmp = idx / CP;
    const int tap = tmp % 9;
    const int o   = tmp / 9;
    wp[idx] = (c < C) ? w[((size_t)o * C + c) * 9 + tap] : 0.0f;
}

// xi[n, cp, hw, i] = (2*cp+i < C) ? x[n, 2*cp+i, hw] : 0
__global__ void pack_input_ilv(const float* __restrict__ x, float* __restrict__ xi,
                               int N, int C, int CP2, int HW)
{
    const int idx = blockIdx.x * blockDim.x + threadIdx.x;
    const int total = N * CP2 * HW * 2;
    if (idx >= total) return;
    const int i   = idx & 1;
    const int hw  = (idx >> 1) % HW;
    const int tmp = (idx >> 1) / HW;
    const int cp  = tmp % CP2;
    const int n   = tmp / CP2;
    const int c   = 2 * cp + i;
    xi[idx] = (c < C) ? x[((size_t)n * C + c) * HW + hw] : 0.0f;
}

extern "C" void kernel_launch(void* const* d_in, const int* in_sizes, int n_in,
                              void* d_out, int out_size, void* d_ws, size_t ws_size,
                              hipStream_t stream)
{
    const float* x  = (const float*)d_in[0];   // [16,3,256,256]
    const float* w1 = (const float*)d_in[1];   // [64,3,3,3]
    const float* w2 = (const float*)d_in[2];   // [128,64,3,3]
    float* out = (float*)d_out;                // [16,128,252,252]

    // Workspace layout (floats):
    //   h   : interleaved [16,32,254,254,2] = 66,064,384
    //   xi  : interleaved [16,2,256,256,2]  =  4,194,304
    //   w1p : [64,9,4]                      =      2,304
    //   w2p : [128,9,64]                    =     73,728
    float* h   = (float*)d_ws;
    float* xi  = h   + (size_t)66064384;
    float* w1p = xi  + (size_t)4194304;
    float* w2p = w1p + (size_t)2304;

    // ---- pack ----
    {
        const int total = 16 * 2 * 256 * 256 * 2;
        pack_input_ilv<<<(total + 255) / 256, 256, 0, stream>>>(x, xi, 16, 3, 2, 256 * 256);
    }
    {
        const int total = 64 * 9 * 4;
        pack_weights<<<(total + 255) / 256, 256, 0, stream>>>(w1, w1p, 64, 3, 4);
    }
    {
        const int total = 128 * 9 * 64;
        pack_weights<<<(total + 255) / 256, 256, 0, stream>>>(w2, w2p, 128, 64, 64);
    }

    // ---- conv1: xi -> h (interleaved out) ----
    {
        constexpr int HIN = 256, WIN = 256, OCN = 64, N = 16;
        constexpr int HO = HIN - 2, WO = WIN - 2;
        constexpr int owTiles = (WO + 31) / 32, ocTiles = OCN / 16;
        const long waves = (long)N * HO * owTiles * ocTiles;   // 130048
        const long blocks = (waves + 7) / 8;
        conv3x3_wmma_f32<4, HIN, WIN, OCN, true>
            <<<dim3((unsigned)blocks), dim3(256), 0, stream>>>(xi, w1p, h, (int)waves);
    }

    // ---- conv2: h -> out (plain NCHW out) ----
    {
        constexpr int HIN = 254, WIN = 254, OCN = 128, N = 16;
        constexpr int HO = HIN - 2, WO = WIN - 2;
        constexpr int owTiles = (WO + 31) / 32, ocTiles = OCN / 16;
        const long waves = (long)N * HO * owTiles * ocTiles;   // 258048
        const long blocks = (waves + 7) / 8;
        conv3x3_wmma_f32<64, HIN, WIN, OCN, false>
            <<<dim3((unsigned)blocks), dim3(256), 0, stream>>>(h, w2p, out, (int)waves);
    }
}